// ConcatPairedNodeAttention_58385785421904
// MI455X (gfx1250) — compile-verified
//
#include <hip/hip_runtime.h>
#include <hip/hip_bf16.h>

typedef __attribute__((ext_vector_type(2))) float v2f;
typedef __attribute__((ext_vector_type(8))) float v8f;

#define IN_DIM   256
#define OUT_DIM  256
#define BATCH    2
#define DEG      16
#define SLOPE    0.2f

// LDS tile: 128 K-pairs x 64 cols x 2 floats = 64 KB, XOR-16 swizzled on pair
// parity so the two wave halves (even/odd pair) hit disjoint bank halves.
#define LDS_PAIR_PITCH 128              // floats per pair-row
__device__ __forceinline__ int widx(int pi, int col) {
    return pi * LDS_PAIR_PITCH + ((col ^ ((pi & 1) << 4)) << 1);
}

// ---------------------------------------------------------------------------
// GEMM with row gather + bias:  out[m, :] = nodes[gather[m/B], m%B, :] @ W + bias
// W is [IN_DIM, OUT_DIM] row-major.  Block = 4 waves sharing one LDS-staged
// 256x64 weight slab; each wave -> 16(M) x 64(N) via V_WMMA_F32_16X16X4_F32.
// ---------------------------------------------------------------------------
__global__ void __launch_bounds__(128)
gemm_gather_bias(const float* __restrict__ nodes,
                 const int*   __restrict__ gather,
                 const float* __restrict__ W,
                 const float* __restrict__ bias,
                 float*       __restrict__ out,
                 int Mrows)
{
    __shared__ float ldsW[128 * LDS_PAIR_PITCH];   // 64 KB

    const int tid  = threadIdx.x;
    const int wave = tid >> 5;
    const int lane = tid & 31;
    const int half = lane >> 4;     // 0: lanes 0-15, 1: lanes 16-31
    const int l15  = lane & 15;

    const int tileM = blockIdx.x * 4 + wave;        // wave-uniform
    const int nbase = blockIdx.y * 64;

    // ---- cooperative stage of W[:, nbase:nbase+64] into LDS (pair layout) ----
    {
        const float4* __restrict__ Wv4 =
            (const float4*)(W + nbase);             // rows of 64 cols, 16 float4s
        #pragma unroll 4
        for (int idx = tid; idx < 256 * 16; idx += 128) {
            const int r  = idx >> 4;                // K row 0..255
            const int c4 = (idx & 15) << 2;         // col 0,4,...,60
            const float4 v = Wv4[(size_t)r * (OUT_DIM / 4) + (c4 >> 2)];
            const int pi = r >> 1, e = r & 1;
            ldsW[widx(pi, c4 + 0) + e] = v.x;
            ldsW[widx(pi, c4 + 1) + e] = v.y;
            ldsW[widx(pi, c4 + 2) + e] = v.z;
            ldsW[widx(pi, c4 + 3) + e] = v.w;
        }
    }
    __syncthreads();

    if (tileM * 16 >= Mrows) return;                // uniform exit, EXEC full

    // Per-lane A row pointer (clamped so EXEC stays all-ones through WMMA).
    int m = tileM * 16 + l15;
    if (m >= Mrows) m = Mrows - 1;
    const int node = gather[m >> 1];                // BATCH == 2
    const float* __restrict__ arow =
        nodes + ((size_t)node * BATCH + (m & 1)) * IN_DIM;

    v8f acc[4] = {v8f{}, v8f{}, v8f{}, v8f{}};

    #pragma unroll 8
    for (int k = 0; k < IN_DIM; k += 4) {
        // A fragment 16x4: VGPR0 = K k+2*half, VGPR1 = K k+2*half+1
        v2f a;
        a.x = arow[k + half * 2];
        a.y = arow[k + half * 2 + 1];
        // B fragments from LDS: pair index = k/2 + half (even/odd by half)
        const int pi = (k >> 1) + half;
        const int sw = (pi & 1) << 4;
        const float* lp = ldsW + pi * LDS_PAIR_PITCH;
        v2f b0 = *(const v2f*)(lp + (((l15 +  0) ^ sw) << 1));
        v2f b1 = *(const v2f*)(lp + (((l15 + 16) ^ sw) << 1));
        v2f b2 = *(const v2f*)(lp + (((l15 + 32) ^ sw) << 1));
        v2f b3 = *(const v2f*)(lp + (((l15 + 48) ^ sw) << 1));
        acc[0] = __builtin_amdgcn_wmma_f32_16x16x4_f32(false, a, false, b0, (short)0, acc[0], false, false);
        acc[1] = __builtin_amdgcn_wmma_f32_16x16x4_f32(false, a, false, b1, (short)0, acc[1], false, false);
        acc[2] = __builtin_amdgcn_wmma_f32_16x16x4_f32(false, a, false, b2, (short)0, acc[2], false, false);
        acc[3] = __builtin_amdgcn_wmma_f32_16x16x4_f32(false, a, false, b3, (short)0, acc[3], false, false);
    }

    // Epilogue: C/D layout -> col = lane&15, row = half*8 + vgpr
    const int mrow0 = tileM * 16 + half * 8;
    #pragma unroll
    for (int s = 0; s < 4; ++s) {
        const int col = nbase + s * 16 + l15;
        const float bv = bias[col];
        #pragma unroll
        for (int r = 0; r < 8; ++r) {
            const int mr = mrow0 + r;
            if (mr < Mrows) out[(size_t)mr * OUT_DIM + col] = acc[s][r] + bv;
        }
    }
}

// ---------------------------------------------------------------------------
// out[r] = sum_o P[r, o] * wa[o]      (wave per row, shuffle reduction)
// ---------------------------------------------------------------------------
__global__ void __launch_bounds__(256)
row_dot(const float* __restrict__ P, const float* __restrict__ wa,
        float* __restrict__ out, int rows)
{
    const int row  = blockIdx.x * 8 + (threadIdx.x >> 5);
    if (row >= rows) return;
    const int lane = threadIdx.x & 31;
    const float* p = P + (size_t)row * OUT_DIM;
    float s = 0.f;
    #pragma unroll
    for (int o = lane; o < OUT_DIM; o += 32) s += p[o] * wa[o];
    #pragma unroll
    for (int off = 16; off > 0; off >>= 1) s += __shfl_xor(s, off, 32);
    if (lane == 0) out[row] = s;
}

// ---------------------------------------------------------------------------
// Per (source, batch): leaky_relu -> clip -> exp over its 16 contiguous edges,
// then normalize in-register (edges sorted by source, DEG each -> atomic-free,
// deterministic).
// ---------------------------------------------------------------------------
__global__ void __launch_bounds__(256)
edge_softmax(const float* __restrict__ a_src, const float* __restrict__ a_tgt,
             const int* __restrict__ sid, const int* __restrict__ tid,
             const float* __restrict__ ba, float* __restrict__ alpha, int Ns)
{
    const int t = blockIdx.x * blockDim.x + threadIdx.x;
    if (t >= Ns * BATCH) return;
    const int s = t >> 1;
    const int b = t & 1;
    const float bias = ba[0];
    float ex[DEG];
    float denom = 0.f;
    #pragma unroll
    for (int j = 0; j < DEG; ++j) {
        const int e = s * DEG + j;
        float sc = a_src[sid[e] * BATCH + b] + a_tgt[tid[e] * BATCH + b] + bias;
        sc = (sc < 0.f) ? sc * SLOPE : sc;          // LeakyReLU
        sc = fminf(fmaxf(sc, -2.f), 2.f);           // clip
        const float v = expf(sc);
        ex[j] = v;
        denom += v;
    }
    #pragma unroll
    for (int j = 0; j < DEG; ++j)
        alpha[(size_t)(s * DEG + j) * BATCH + b] = ex[j] / denom;
}

// ---------------------------------------------------------------------------
// Faithful aggregation:  gathered[e] = weighted[tid[e]]  (edge-axis index!)
// out[s,b,o] = src_proj[s,b,o] + sum_j alpha[f_j,b] * vals[tid[f_j], b, o]
//   where f_j = tid[s*DEG + j].   Block per source; indirections staged in LDS.
// ---------------------------------------------------------------------------
__global__ void __launch_bounds__(256)
aggregate(const float* __restrict__ src_proj, const float* __restrict__ vals,
          const float* __restrict__ alpha, const int* __restrict__ tid,
          float* __restrict__ out)
{
    __shared__ int   rowv[DEG];
    __shared__ float al[DEG * BATCH];
    const int s = blockIdx.x;
    if (threadIdx.x < DEG) {
        const int f = tid[s * DEG + threadIdx.x];   // edge index (reference quirk)
        rowv[threadIdx.x] = tid[f];                 // value row = tid[f]
        al[threadIdx.x * BATCH + 0] = alpha[(size_t)f * BATCH + 0];
        al[threadIdx.x * BATCH + 1] = alpha[(size_t)f * BATCH + 1];
    }
    __syncthreads();
    #pragma unroll
    for (int i = 0; i < (BATCH * OUT_DIM) / 256; ++i) {
        const int idx = threadIdx.x + i * 256;      // 0 .. 511
        const int b = idx >> 8;
        const int o = idx & 255;
        const size_t obase = ((size_t)s * BATCH + b) * OUT_DIM + o;
        float acc = src_proj[obase];
        #pragma unroll
        for (int j = 0; j < DEG; ++j)
            acc += al[j * BATCH + b] *
                   vals[((size_t)rowv[j] * BATCH + b) * OUT_DIM + o];
        out[obase] = acc;
    }
}

// ---------------------------------------------------------------------------
extern "C" void kernel_launch(void* const* d_in, const int* in_sizes, int n_in,
                              void* d_out, int out_size, void* d_ws, size_t ws_size,
                              hipStream_t stream)
{
    const float* nodes = (const float*)d_in[0];
    const float* Ws    = (const float*)d_in[1];
    const float* bs    = (const float*)d_in[2];
    const float* Wt    = (const float*)d_in[3];
    const float* bt    = (const float*)d_in[4];
    const float* Wv    = (const float*)d_in[5];
    const float* bv    = (const float*)d_in[6];
    const float* Wa    = (const float*)d_in[7];   // [2*OUT_DIM]
    const float* ba    = (const float*)d_in[8];
    const int* usrc    = (const int*)d_in[9];
    const int* utgt    = (const int*)d_in[10];
    const int* sid     = (const int*)d_in[11];
    const int* tid     = (const int*)d_in[12];

    const int Ns = in_sizes[9];
    const int Nt = in_sizes[10];
    const int Ne = in_sizes[11];
    const int Msrc = Ns * BATCH;
    const int Mtgt = Nt * BATCH;

    // Workspace carve-out (256B aligned).
    char* ws = (char*)d_ws;
    size_t off = 0;
    auto alloc = [&](size_t bytes) -> void* {
        void* p = ws + off;
        off += (bytes + 255) & ~(size_t)255;
        return p;
    };
    float* src_proj = (float*)alloc((size_t)Msrc * OUT_DIM * sizeof(float));
    float* tgt_proj = (float*)alloc((size_t)Mtgt * OUT_DIM * sizeof(float));
    float* valsbuf  = (float*)alloc((size_t)Mtgt * OUT_DIM * sizeof(float));
    float* a_src    = (float*)alloc((size_t)Msrc * sizeof(float));
    float* a_tgt    = (float*)alloc((size_t)Mtgt * sizeof(float));
    float* alpha    = (float*)alloc((size_t)Ne * BATCH * sizeof(float));
    (void)ws_size; (void)n_in; (void)out_size;

    const dim3 blk(128);
    auto gemm_grid = [](int M) {
        const int mtiles = (M + 15) / 16;
        return dim3((unsigned)((mtiles + 3) / 4), OUT_DIM / 64);
    };

    gemm_gather_bias<<<gemm_grid(Msrc), blk, 0, stream>>>(nodes, usrc, Ws, bs, src_proj, Msrc);
    gemm_gather_bias<<<gemm_grid(Mtgt), blk, 0, stream>>>(nodes, utgt, Wt, bt, tgt_proj, Mtgt);
    gemm_gather_bias<<<gemm_grid(Mtgt), blk, 0, stream>>>(nodes, utgt, Wv, bv, valsbuf,  Mtgt);

    row_dot<<<(Msrc + 7) / 8, 256, 0, stream>>>(src_proj, Wa,           a_src, Msrc);
    row_dot<<<(Mtgt + 7) / 8, 256, 0, stream>>>(tgt_proj, Wa + OUT_DIM, a_tgt, Mtgt);

    edge_softmax<<<(Ns * BATCH + 255) / 256, 256, 0, stream>>>(a_src, a_tgt, sid, tid, ba, alpha, Ns);

    aggregate<<<Ns, 256, 0, stream>>>(src_proj, valsbuf, alpha, tid, (float*)d_out);
}